// GLogicalLayer_63204738728141
// MI455X (gfx1250) — compile-verified
//
#include <hip/hip_runtime.h>
#include <math.h>

// ---------------- problem constants (from reference) ----------------
constexpr int kD      = 128;
constexpr int kL      = 3;
constexpr int kBATCH  = 128;
constexpr int kRELROW = 401;     // 2R+1
constexpr int kNUMENT = 40000;
constexpr int kU      = 300000;
__constant__ int c_dummy; // keep TU non-trivial

static const int kE[3] = {500000, 1000000, 1500000};
static const int kN[3] = {50000, 200000, 400000};

typedef __attribute__((ext_vector_type(2))) float v2f;
typedef __attribute__((ext_vector_type(8))) float v8f;

// ---------------- utility: zero a float buffer ----------------
__global__ void zero_kernel(float* __restrict__ p, long long n) {
    long long i = (long long)blockIdx.x * blockDim.x + threadIdx.x;
    long long stride = (long long)gridDim.x * blockDim.x;
    for (; i < n; i += stride) p[i] = 0.0f;
}

// ---------------- gather query-relation embedding rows ----------------
// qr[b][:] = rel_emb_l[query_rel[b]+1][:]
__global__ void gather_q_kernel(const float* __restrict__ rel_l,
                                const int* __restrict__ query_rel,
                                float* __restrict__ qr) {
    int b = blockIdx.x;            // 0..127
    int c = threadIdx.x;           // 0..127
    int row = query_rel[b] + 1;
    qr[b * kD + c] = rel_l[(size_t)row * kD + c];
}

// ---------------- f32 WMMA GEMM:  C[M x 128] = A[M x 128] @ W^T (+bias) ----
// One wave computes one 16x16 output tile via V_WMMA_F32_16X16X4_F32.
// block = (32, 8): threadIdx.y selects the 16-col tile (8 tiles cover 128).
__global__ void gemm_nt_wmma(const float* __restrict__ A,
                             const float* __restrict__ W,
                             const float* __restrict__ bias,   // may be null
                             float* __restrict__ C, int M) {
    const int lane    = threadIdx.x;        // 0..31
    const int l16     = lane & 15;
    const int half    = lane >> 4;          // 0 or 1
    const int m0      = blockIdx.x * 16;
    const int j0      = threadIdx.y * 16;

    const int arow    = m0 + l16;
    const int arow_cl = arow < M ? arow : (M - 1);  // clamp; masked on store
    const int kl      = half * 2;                   // K offset this lane covers

    const float* __restrict__ Arow = A + (size_t)arow_cl * kD;
    const float* __restrict__ Wrow = W + (size_t)(j0 + l16) * kD;  // B[k][j]=W[j][k]

    v8f acc = {};
#pragma unroll
    for (int k0 = 0; k0 < kD; k0 += 4) {
        v2f a, b;
        a.x = Arow[k0 + kl];
        a.y = Arow[k0 + kl + 1];
        b.x = Wrow[k0 + kl];
        b.y = Wrow[k0 + kl + 1];
        acc = __builtin_amdgcn_wmma_f32_16x16x4_f32(
            /*neg_a=*/false, a, /*neg_b=*/false, b,
            /*c_mod=*/(short)0, acc, /*reuse_a=*/false, /*reuse_b=*/false);
    }

    const float bj = bias ? bias[j0 + l16] : 0.0f;
#pragma unroll
    for (int v = 0; v < 8; ++v) {
        int m = m0 + v + 8 * half;            // C/D layout: VGPR v, lanes 16-31 -> M+8
        if (m < M) C[(size_t)m * kD + j0 + l16] = acc[v] + bj;
    }
}

// ---------------- per-edge message + scatter-add ----------------
// One wave32 per edge; lane handles 4 contiguous columns (float4).
__global__ void edge_kernel(const int* __restrict__ facts,
                            const int* __restrict__ tidx, int E,
                            const float* __restrict__ hidden,
                            const float* __restrict__ hproj,
                            const float* __restrict__ rproj,
                            const float* __restrict__ qproj,
                            const float* __restrict__ rel_l,
                            const float* __restrict__ walpha_l,
                            const float* __restrict__ balpha_l,
                            float* __restrict__ agg) {
    const int lane = threadIdx.x & 31;
    const long long e = (long long)blockIdx.x * (blockDim.x >> 5) + (threadIdx.x >> 5);
    if (e >= E) return;                       // wave-uniform exit

    const int* f = facts + e * 6;
    const int b = f[0];
    const int n = f[1];
    const int r = f[3] + 1;
    const int t = tidx[e];
    const int c = lane << 2;

    const float4 h4  = *(const float4*)(hidden + (size_t)n * kD + c);
    const float4 hr4 = *(const float4*)(rel_l  + (size_t)r * kD + c);
    const float4 hp4 = *(const float4*)(hproj  + (size_t)n * kD + c);
    const float4 rp4 = *(const float4*)(rproj  + (size_t)r * kD + c);
    const float4 qp4 = *(const float4*)(qproj  + (size_t)b * kD + c);
    const float4 wa4 = *(const float4*)(walpha_l + c);

    float px = hp4.x + rp4.x + qp4.x;
    float py = hp4.y + rp4.y + qp4.y;
    float pz = hp4.z + rp4.z + qp4.z;
    float pw = hp4.w + rp4.w + qp4.w;

    float local = fmaxf(px, 0.0f) * wa4.x + fmaxf(py, 0.0f) * wa4.y
                + fmaxf(pz, 0.0f) * wa4.z + fmaxf(pw, 0.0f) * wa4.w;
#pragma unroll
    for (int off = 16; off > 0; off >>= 1) local += __shfl_xor(local, off);

    const float alpha = 1.0f / (1.0f + __expf(-(local + balpha_l[0])));

    float* ag = agg + (size_t)t * kD + c;
    unsafeAtomicAdd(ag + 0, alpha * (h4.x + hr4.x));
    unsafeAtomicAdd(ag + 1, alpha * (h4.y + hr4.y));
    unsafeAtomicAdd(ag + 2, alpha * (h4.z + hr4.z));
    unsafeAtomicAdd(ag + 3, alpha * (h4.w + hr4.w));
}

// ---------------- final: scores_u[fti[i]] += dot(hidden[i], Wfinal) --------
__global__ void final_dot_kernel(const float* __restrict__ hidden,
                                 const float* __restrict__ Wfinal,
                                 const int* __restrict__ fti, int N2,
                                 float* __restrict__ scores_u) {
    const int lane = threadIdx.x & 31;
    const long long i = (long long)blockIdx.x * (blockDim.x >> 5) + (threadIdx.x >> 5);
    if (i >= N2) return;
    const int c = lane << 2;
    const float4 h4 = *(const float4*)(hidden + (size_t)i * kD + c);
    const float4 w4 = *(const float4*)(Wfinal + c);
    float local = h4.x * w4.x + h4.y * w4.y + h4.z * w4.z + h4.w * w4.w;
#pragma unroll
    for (int off = 16; off > 0; off >>= 1) local += __shfl_xor(local, off);
    if (lane == 0) unsafeAtomicAdd(&scores_u[fti[i]], local);
}

// ---------------- scatter scores into (BATCH, NUM_ENT) ----------------
__global__ void scatter_kernel(const int* __restrict__ tail_ents,
                               const float* __restrict__ scores_u,
                               float* __restrict__ out) {
    int u = blockIdx.x * blockDim.x + threadIdx.x;
    if (u >= kU) return;
    int b = tail_ents[2 * u];
    int en = tail_ents[2 * u + 1];
    out[(size_t)b * kNUMENT + en] = scores_u[u];
}

// ---------------------------------------------------------------------------
extern "C" void kernel_launch(void* const* d_in, const int* in_sizes, int n_in,
                              void* d_out, int out_size, void* d_ws, size_t ws_size,
                              hipStream_t stream) {
    const float* rel_emb   = (const float*)d_in[0];   // L x 401 x 128
    const float* Ws        = (const float*)d_in[1];   // L x 128 x 128
    const float* Wr        = (const float*)d_in[2];
    const float* Wqr       = (const float*)d_in[3];
    const float* bqr       = (const float*)d_in[4];   // L x 128
    const float* walpha    = (const float*)d_in[5];   // L x 128
    const float* balpha    = (const float*)d_in[6];   // L
    const float* Wh        = (const float*)d_in[7];   // L x 128 x 128
    const float* Wfinal    = (const float*)d_in[8];   // 128
    const int*   query_rel = (const int*)d_in[9];     // 128
    const int*   facts[3]  = {(const int*)d_in[10], (const int*)d_in[11], (const int*)d_in[12]};
    const int*   tidx[3]   = {(const int*)d_in[13], (const int*)d_in[14], (const int*)d_in[15]};
    const int*   tail_ents = (const int*)d_in[16];    // U x 2
    const int*   fti       = (const int*)d_in[17];    // N2
    float* out = (float*)d_out;

    // workspace layout (floats)
    float* ws = (float*)d_ws;
    float* hidden   = ws;                                   // 400000*128
    float* agg      = hidden + (size_t)400000 * kD;         // 400000*128
    float* hproj    = agg    + (size_t)400000 * kD;         // 200000*128
    float* qr       = hproj  + (size_t)200000 * kD;         // 128*128
    float* qproj    = qr     + (size_t)kBATCH * kD;         // 128*128
    float* rproj    = qproj  + (size_t)kBATCH * kD;         // 401*128
    float* scores_u = rproj  + (size_t)kRELROW * kD;        // 300000

    const dim3 gblk(32, 8);

    // hidden starts as zeros (BATCH x D)
    zero_kernel<<<64, 256, 0, stream>>>(hidden, (long long)kBATCH * kD);

    for (int l = 0; l < kL; ++l) {
        const int Nprev = (l == 0) ? kBATCH : kN[l - 1];
        const int Nl = kN[l];
        const int El = kE[l];
        const float* rel_l = rel_emb + (size_t)l * kRELROW * kD;

        // hproj = hidden @ Ws[l]^T
        gemm_nt_wmma<<<dim3((Nprev + 15) / 16), gblk, 0, stream>>>(
            hidden, Ws + (size_t)l * kD * kD, nullptr, hproj, Nprev);
        // rproj = rel_emb[l] @ Wr[l]^T
        gemm_nt_wmma<<<dim3((kRELROW + 15) / 16), gblk, 0, stream>>>(
            rel_l, Wr + (size_t)l * kD * kD, nullptr, rproj, kRELROW);
        // qr[b] = rel_emb[l][query_rel[b]+1] ; qproj = qr @ Wqr[l]^T + bqr[l]
        gather_q_kernel<<<kBATCH, kD, 0, stream>>>(rel_l, query_rel, qr);
        gemm_nt_wmma<<<dim3(kBATCH / 16), gblk, 0, stream>>>(
            qr, Wqr + (size_t)l * kD * kD, bqr + (size_t)l * kD, qproj, kBATCH);

        // agg = 0 ; edge scatter-add
        zero_kernel<<<2048, 256, 0, stream>>>(agg, (long long)Nl * kD);
        edge_kernel<<<(El + 7) / 8, 256, 0, stream>>>(
            facts[l], tidx[l], El, hidden, hproj, rproj, qproj,
            rel_l, walpha + (size_t)l * kD, balpha + l, agg);

        // hidden = agg @ Wh[l]^T
        gemm_nt_wmma<<<dim3((Nl + 15) / 16), gblk, 0, stream>>>(
            agg, Wh + (size_t)l * kD * kD, nullptr, hidden, Nl);
    }

    // scores_u[fti[i]] += dot(hidden[i], Wfinal)
    zero_kernel<<<512, 256, 0, stream>>>(scores_u, (long long)kU);
    final_dot_kernel<<<(kN[2] + 7) / 8, 256, 0, stream>>>(
        hidden, Wfinal, fti, kN[2], scores_u);

    // scatter into (BATCH, NUM_ENT)
    zero_kernel<<<4096, 256, 0, stream>>>(out, (long long)kBATCH * kNUMENT);
    scatter_kernel<<<(kU + 255) / 256, 256, 0, stream>>>(tail_ents, scores_u, out);
}